// Point_SACgai_66700842107339
// MI455X (gfx1250) — compile-verified
//
#include <hip/hip_runtime.h>
#include <hip/hip_bf16.h>

#define B_ 16
#define C_ 256
#define N_ 2048
#define BN_EPS 1e-5f

typedef __attribute__((ext_vector_type(2))) float v2f;
typedef __attribute__((ext_vector_type(8))) float v8f;
typedef __attribute__((ext_vector_type(4))) unsigned int u32x4;
typedef __attribute__((ext_vector_type(8))) int i32x8;
typedef __attribute__((ext_vector_type(4))) int i32x4;

// ---------------------------------------------------------------------------
// WMMA f32 16x16x4.  ISA layouts (wave32):
//  A (16x4, MxK): lane<16 -> M=lane, elems K=k0+0,k0+1 ; lane>=16 -> K=k0+2,k0+3
//  B (4x16, KxN): same K pattern per lane-half, N = lane&15
//  C/D (16x16):   VGPR r: lanes 0-15 -> M=r, lanes 16-31 -> M=r+8, N = lane&15
// ---------------------------------------------------------------------------
__device__ __forceinline__ v8f wmma4(v2f a, v2f b, v8f c) {
  return __builtin_amdgcn_wmma_f32_16x16x4_f32(false, a, false, b, (short)0, c,
                                               false, false);
}

__device__ __forceinline__ float wredmax(float v) {
  #pragma unroll
  for (int o = 16; o > 0; o >>= 1) v = fmaxf(v, __shfl_down(v, o));
  return v;
}
__device__ __forceinline__ float wredsum(float v) {
  #pragma unroll
  for (int o = 16; o > 0; o >>= 1) v += __shfl_down(v, o);
  return v;
}

// ---------------------------------------------------------------------------
// TDM: stage a 32-row x 64-col f32 tile (row stride N_ elems) from global
// into LDS at byte offset lds_byte, as ONE tensor_load_to_lds descriptor.
// D# group0: count=1 | lds_addr | global_addr[56:0] | type=2
// D# group1: data_size=4B, tensor_dim0=64, tensor_dim1=32, tile=64x32,
//            tensor_dim0_stride=N_.  Groups 2/3 unused (zeros, tile_dim3=0).
// Tracked with TENSORcnt.
// ---------------------------------------------------------------------------
__device__ __forceinline__ void tdm_stage_b(unsigned lds_byte,
                                            const float* gptr) {
  const unsigned long long ga = (unsigned long long)(uintptr_t)gptr;
  u32x4 g0;
  g0[0] = 1u;                                                  // count=1
  g0[1] = lds_byte;                                            // lds_addr
  g0[2] = (unsigned)ga;                                        // gaddr[31:0]
  g0[3] = (unsigned)((ga >> 32) & 0x01ffffffu) | 0x80000000u;  // [56:32]|type2
  i32x8 g1;
  g1[0] = 0x00020000;        // data_size=2 (4 bytes); no multicast/pad/iter
  g1[1] = (int)(64u << 16);  // tensor_dim0[15:0] = 64
  g1[2] = (int)(32u << 16);  // tensor_dim0[31:16]=0, tensor_dim1[15:0]=32
  g1[3] = (int)(64u << 16);  // tensor_dim1[31:16]=0, tile_dim0=64
  g1[4] = 32;                // tile_dim1=32, tile_dim2=0 (unused)
  g1[5] = N_;                // tensor_dim0_stride[31:0] = 2048 elems
  g1[6] = 0;                 // stride[47:32]=0, tensor_dim1_stride[15:0]=0
  g1[7] = 0;
  const i32x4 z4 = {};
  const i32x8 z8 = {};
  __builtin_amdgcn_tensor_load_to_lds(g0, g1, z4, z4, z8, 0);
}

// ---------------------------------------------------------------------------
// Double-buffered GEMM mainloop. Block tile = 128(M) x 64(N); 8 waves, each
// wave owns a 16x64 slice (4 accumulators). The shared B chunk (32K x 64N
// f32 = 8KB) is staged into LDS by the Tensor Data Mover (one descriptor per
// stage, issued by wave 0), double buffered via s_wait_tensorcnt + barriers.
// AFetch(k) returns the lane's A operand pair for K = k .. k+1.
// ---------------------------------------------------------------------------
template <typename AFetch>
__device__ __forceinline__ void gemm_mainloop(AFetch afetch,
                                              const float* __restrict__ Bg,
                                              int Ktot, float* sm, v8f& a0,
                                              v8f& a1, v8f& a2, v8f& a3) {
  const int lane = threadIdx.x & 31;
  const int half = lane >> 4, l16 = lane & 15;
  const unsigned smbase = (unsigned)(uintptr_t)sm;  // low 32 bits = LDS offset
  const int S = Ktot >> 5;  // stages of K=32

  if (threadIdx.x < 32) tdm_stage_b(smbase, Bg);
  for (int s = 0; s < S; ++s) {
    const int cur = s & 1;
    if (s + 1 < S) {
      if (threadIdx.x < 32)
        tdm_stage_b(smbase + (unsigned)((cur ^ 1) * 8192),
                    Bg + (size_t)(s + 1) * 32 * N_);
      __builtin_amdgcn_s_wait_tensorcnt(1);  // my stage-s TDM done
    } else {
      __builtin_amdgcn_s_wait_tensorcnt(0);
    }
    __syncthreads();  // stage-s tile visible to all waves
    const float* bb = sm + cur * 2048;
    #pragma unroll
    for (int g = 0; g < 8; ++g) {
      const int k = g * 4 + 2 * half;
      v2f av = afetch((s << 5) + k);
      const float* bp = bb + k * 64 + l16;
      v2f b0, b1, b2, b3;
      b0.x = bp[0];  b0.y = bp[64];
      b1.x = bp[16]; b1.y = bp[80];
      b2.x = bp[32]; b2.y = bp[96];
      b3.x = bp[48]; b3.y = bp[112];
      a0 = wmma4(av, b0, a0);
      a1 = wmma4(av, b1, a1);
      a2 = wmma4(av, b2, a2);
      a3 = wmma4(av, b3, a3);
    }
    __syncthreads();  // readers done before buffer is re-filled
  }
}

// ---------------------------------------------------------------------------
// Y[b][o][n] = sum_c W[o][c] * X[b][c][n] + bias[o]
// block tile 128(o) x 64(n): 64 blocks/batch * 16 = 1024 blocks of 256 thr.
// ---------------------------------------------------------------------------
__global__ void k_conv1x1(const float* __restrict__ W,
                          const float* __restrict__ bias,
                          const float* __restrict__ X, float* __restrict__ Y) {
  __shared__ float sm[2 * 2048];
  const int wid = threadIdx.x >> 5;
  const int lane = threadIdx.x & 31;
  const int half = lane >> 4, l16 = lane & 15;
  const int b = blockIdx.x >> 6;
  const int t = blockIdx.x & 63;
  const int m0 = (t >> 5) * 128 + wid * 16;
  const int n0 = (t & 31) * 64;
  const float* Xb = X + (size_t)b * C_ * N_;
  const float* ap = W + (size_t)(m0 + l16) * C_;

  v8f a0 = {}, a1 = {}, a2 = {}, a3 = {};
  gemm_mainloop([&](int k) { return *(const v2f*)(ap + k); },
                Xb + n0, C_, sm, a0, a1, a2, a3);

  #pragma unroll
  for (int r = 0; r < 8; ++r) {
    const int m = m0 + r + 8 * half;
    const float bv = bias[m];
    float* yp = Y + ((size_t)b * C_ + m) * N_ + n0 + l16;
    yp[0]  = a0[r] + bv;
    yp[16] = a1[r] + bv;
    yp[32] = a2[r] + bv;
    yp[48] = a3[r] + bv;
  }
}

// ---------------------------------------------------------------------------
// E[i][j] = sum_c X[b][c][i] * X0[b][c][j]   (one batch)
// block tile 128(i) x 64(j): 16*32 = 512 blocks.
// ---------------------------------------------------------------------------
__global__ void k_energy(const float* __restrict__ X,
                         const float* __restrict__ X0, float* __restrict__ E,
                         int b) {
  __shared__ float sm[2 * 2048];
  const int wid = threadIdx.x >> 5;
  const int lane = threadIdx.x & 31;
  const int half = lane >> 4, l16 = lane & 15;
  const int i0 = (blockIdx.x >> 5) * 128 + wid * 16;
  const int j0 = (blockIdx.x & 31) * 64;
  const float* Xb = X + (size_t)b * C_ * N_;
  const float* Zb = X0 + (size_t)b * C_ * N_;
  const float* ap = Xb + i0 + l16;  // A[m][k] = X[k*N + i0+m], K-stride = N

  v8f a0 = {}, a1 = {}, a2 = {}, a3 = {};
  gemm_mainloop(
      [&](int k) {
        v2f a;
        a.x = ap[(size_t)k * N_];
        a.y = ap[(size_t)k * N_ + N_];
        return a;
      },
      Zb + j0, C_, sm, a0, a1, a2, a3);

  #pragma unroll
  for (int r = 0; r < 8; ++r) {
    float* ep = E + (size_t)(i0 + r + 8 * half) * N_ + j0 + l16;
    ep[0]  = a0[r];
    ep[16] = a1[r];
    ep[32] = a2[r];
    ep[48] = a3[r];
  }
}

// ---------------------------------------------------------------------------
// Row softmax in place. One block (256 thr) per row of 2048.
// ---------------------------------------------------------------------------
__global__ void k_softmax(float* __restrict__ E) {
  float* row = E + (size_t)blockIdx.x * N_;
  const int wid = threadIdx.x >> 5;
  const int lane = threadIdx.x & 31;
  __shared__ float sh[8];

  float m = -1e30f;
  for (int j = threadIdx.x; j < N_; j += 256) m = fmaxf(m, row[j]);
  m = wredmax(m);
  if (lane == 0) sh[wid] = m;
  __syncthreads();
  if (wid == 0) {
    float v = (lane < 8) ? sh[lane] : -1e30f;
    v = wredmax(v);
    if (lane == 0) sh[0] = v;
  }
  __syncthreads();
  m = sh[0];
  __syncthreads();

  float s = 0.f;
  for (int j = threadIdx.x; j < N_; j += 256) {
    const float e = expf(row[j] - m);
    row[j] = e;
    s += e;
  }
  s = wredsum(s);
  if (lane == 0) sh[wid] = s;
  __syncthreads();
  if (wid == 0) {
    float v = (lane < 8) ? sh[lane] : 0.f;
    v = wredsum(v);
    if (lane == 0) sh[0] = v;
  }
  __syncthreads();
  const float inv = 1.f / sh[0];
  for (int j = threadIdx.x; j < N_; j += 256) row[j] *= inv;
}

__global__ void k_zero(float* __restrict__ p, int n) {
  const int i = blockIdx.x * 256 + threadIdx.x;
  if (i < n) p[i] = 0.f;
}

// colsum[j] = sum_i A[i][j].  32 blocks: 8 column groups x 4 row chunks.
__global__ void k_colsum(const float* __restrict__ A,
                         float* __restrict__ colsum) {
  const int j = (blockIdx.x & 7) * 256 + threadIdx.x;
  const int r0 = (blockIdx.x >> 3) * 512;
  float s = 0.f;
  for (int i = r0; i < r0 + 512; ++i) s += A[(size_t)i * N_ + j];
  atomicAdd(&colsum[j], s);
}

// ---------------------------------------------------------------------------
// D[b][c][m] = X[b][c][m] - (sum_n X0[b][c][n]*A[n][m]) / (1e-9 + colsum[m])
// block tile 128(c) x 64(m): 2*32 = 64 blocks. K = N = 2048 (64 stages).
// ---------------------------------------------------------------------------
__global__ void k_x2d(const float* __restrict__ X0, const float* __restrict__ A,
                      const float* __restrict__ colsum,
                      const float* __restrict__ X, float* __restrict__ D,
                      int b) {
  __shared__ float sm[2 * 2048];
  const int wid = threadIdx.x >> 5;
  const int lane = threadIdx.x & 31;
  const int half = lane >> 4, l16 = lane & 15;
  const int c0 = (blockIdx.x >> 5) * 128 + wid * 16;
  const int m0 = (blockIdx.x & 31) * 64;
  const float* Zb = X0 + (size_t)b * C_ * N_;
  const float* ap = Zb + (size_t)(c0 + l16) * N_;  // contiguous along K

  v8f a0 = {}, a1 = {}, a2 = {}, a3 = {};
  gemm_mainloop([&](int k) { return *(const v2f*)(ap + k); },
                A + m0, N_, sm, a0, a1, a2, a3);

  #pragma unroll
  for (int r = 0; r < 8; ++r) {
    const int c = c0 + r + 8 * half;
    const float* xp = X + ((size_t)b * C_ + c) * N_ + m0 + l16;
    float* dp = D + ((size_t)b * C_ + c) * N_ + m0 + l16;
    #pragma unroll
    for (int s = 0; s < 4; ++s) {
      const int m = m0 + s * 16 + l16;
      const float sc = 1.f / (1e-9f + colsum[m]);
      const float acc =
          (s == 0) ? a0[r] : (s == 1) ? a1[r] : (s == 2) ? a2[r] : a3[r];
      dp[s * 16] = xp[s * 16] - acc * sc;
    }
  }
}

// One block per channel: mean & var of T over (b, n).
__global__ void k_bnstats(const float* __restrict__ T,
                          float* __restrict__ stats) {
  const int c = blockIdx.x;
  const int wid = threadIdx.x >> 5;
  const int lane = threadIdx.x & 31;
  __shared__ float sh[16];
  float s = 0.f, s2 = 0.f;
  for (int idx = threadIdx.x; idx < B_ * N_; idx += 256) {
    const int bb = idx >> 11;
    const int n = idx & (N_ - 1);
    const float v = T[((size_t)bb * C_ + c) * N_ + n];
    s += v;
    s2 += v * v;
  }
  s = wredsum(s);
  s2 = wredsum(s2);
  if (lane == 0) { sh[wid] = s; sh[8 + wid] = s2; }
  __syncthreads();
  if (threadIdx.x == 0) {
    float ts = 0.f, ts2 = 0.f;
    #pragma unroll
    for (int i = 0; i < 8; ++i) { ts += sh[i]; ts2 += sh[8 + i]; }
    const float inv = 1.f / (float)(B_ * N_);
    const float mean = ts * inv;
    stats[c] = mean;
    stats[C_ + c] = ts2 * inv - mean * mean;  // population variance (ddof=0)
  }
}

// out = x + alpha * relu((t - mean)*rsqrt(var+eps)*gamma + bnbeta) + beta
__global__ void k_final(const float* __restrict__ X, const float* __restrict__ T,
                        const float* __restrict__ stats,
                        const float* __restrict__ gamma,
                        const float* __restrict__ bnbeta,
                        const float* __restrict__ alpha,
                        const float* __restrict__ beta,
                        float* __restrict__ out) {
  const size_t idx = (size_t)blockIdx.x * 256 + threadIdx.x;
  const int c = (int)((idx >> 11) & (C_ - 1));
  const float mean = stats[c];
  const float var = stats[C_ + c];
  const float th = (T[idx] - mean) * rsqrtf(var + BN_EPS);
  const float xr = fmaxf(th * gamma[c] + bnbeta[c], 0.f);
  out[idx] = X[idx] + alpha[c] * xr + beta[c];
}

// ---------------------------------------------------------------------------
extern "C" void kernel_launch(void* const* d_in, const int* in_sizes, int n_in,
                              void* d_out, int out_size, void* d_ws,
                              size_t ws_size, hipStream_t stream) {
  (void)in_sizes; (void)n_in; (void)out_size; (void)ws_size;
  const float* x      = (const float*)d_in[0];
  // d_in[1] = p, unused by reference forward
  const float* v_w    = (const float*)d_in[2];
  const float* v_b    = (const float*)d_in[3];
  const float* t_w    = (const float*)d_in[4];
  const float* t_b    = (const float*)d_in[5];
  const float* gamma  = (const float*)d_in[6];
  const float* bnbeta = (const float*)d_in[7];
  const float* alpha  = (const float*)d_in[8];
  const float* beta   = (const float*)d_in[9];
  float* out = (float*)d_out;

  float* ws = (float*)d_ws;
  const size_t BCN = (size_t)B_ * C_ * N_;   // 8388608
  float* x0     = ws;                         // [B,C,N]   (reused as t later)
  float* dbuf   = ws + BCN;                   // [B,C,N]
  float* attn   = ws + 2 * BCN;               // [N,N] per-batch scratch
  float* colsum = attn + (size_t)N_ * N_;     // [N]
  float* stats  = colsum + N_;                // [2*C] mean, var

  // x0 = v_w @ x + v_b
  k_conv1x1<<<1024, 256, 0, stream>>>(v_w, v_b, x, x0);

  for (int b = 0; b < B_; ++b) {
    k_energy<<<512, 256, 0, stream>>>(x, x0, attn, b);
    k_softmax<<<2048, 256, 0, stream>>>(attn);
    k_zero<<<8, 256, 0, stream>>>(colsum, N_);
    k_colsum<<<32, 256, 0, stream>>>(attn, colsum);
    k_x2d<<<64, 256, 0, stream>>>(x0, attn, colsum, x, dbuf, b);
  }

  // t = t_w @ d + t_b   (t overwrites x0 buffer, which is dead now)
  k_conv1x1<<<1024, 256, 0, stream>>>(t_w, t_b, dbuf, x0);
  k_bnstats<<<256, 256, 0, stream>>>(x0, stats);
  k_final<<<32768, 256, 0, stream>>>(x, x0, stats, gamma, bnbeta, alpha, beta,
                                     out);
}